// EAHead_22290880266617
// MI455X (gfx1250) — compile-verified
//
#include <hip/hip_runtime.h>
#include <hip/hip_bf16.h>

// External-Attention head for MI455X (gfx1250, wave32, WMMA).
//
// Algebraic fusion: y = Wf@x+bf is only consumed by the k-projection and the
// residual uses the original x, so
//   logits = (Wk@Wf) @ x + (Wk@bf)      -- 68.7 GFLOP collapses to 8.6 GFLOP
//   out    = Wv @ knorm(softmax_N(logits)) + x
// ~1.7 GB total traffic -> ~75 us at 23.3 TB/s. The two GEMMs need ~300
// TFLOP/s f32 to stay BW-bound -> V_WMMA_F32_16X16X4_F32 (full f32; softmax
// logits are too rounding-sensitive for bf16 inputs).
//
// This revision widens every wave to a 64-column strip with the column remap
// n = n0 + j*4 + t (tile t, B-column j): one b128 global load per lane feeds
// FOUR 16x16 N-tiles, and all global loads/stores become contiguous b128
// (512 B per wave per instruction) -- 4x fewer VMEM instructions per WMMA.

#define Cdim 256
#define Kdim 32
#define Ndim 65536   // 256*256 spatial
#define Bdim 8

typedef float v2f __attribute__((ext_vector_type(2)));
typedef float v4f __attribute__((ext_vector_type(4)));
typedef float v8f __attribute__((ext_vector_type(8)));

#define WMMA_F32(a, b, c) \
    __builtin_amdgcn_wmma_f32_16x16x4_f32(false, (a), false, (b), (short)0, (c), false, false)

// ---------------------------------------------------------------------------
// K0: fold Wkf = Wk @ Wf  (32x256), bk = Wk @ bf (32).  Tiny: one block.
// ---------------------------------------------------------------------------
__global__ __launch_bounds__(256) void k0_fold(const float* __restrict__ Wk,
                                               const float* __restrict__ Wf,
                                               const float* __restrict__ bf,
                                               float* __restrict__ Wkf,
                                               float* __restrict__ bk) {
    const int i = threadIdx.x;                  // output column (coalesced over Wf rows)
    for (int k = 0; k < Kdim; ++k) {
        float acc = 0.f;
        for (int c = 0; c < Cdim; ++c)
            acc = fmaf(Wk[k * Cdim + c], Wf[c * Cdim + i], acc);
        Wkf[k * Cdim + i] = acc;
    }
    if (i < Kdim) {
        float acc = 0.f;
        for (int c = 0; c < Cdim; ++c)
            acc = fmaf(Wk[i * Cdim + c], bf[c], acc);
        bk[i] = acc;
    }
}

// ---------------------------------------------------------------------------
// K1: L[b,k,n] = sum_c Wkf[k,c] * x[b,c,n] + bk[k]
// 8 waves/block; each wave: 32(M) x 64(N), K=256 in steps of 4.
// B-tile t, column j  <->  global n = n0 + j*4 + t, so each lane's b128 load
// of x[row][n0+lanelo*4 .. +3] supplies all 4 N-tiles for that row.
// A 16x4 layout (ISA 7.12.2): lanes 0-15 M=0..15; VGPR0 K = c0+2*half,
// VGPR1 K = c0+2*half+1.  C/D: VGPR r -> row r+8*half, col = lane%16.
// ---------------------------------------------------------------------------
__global__ __launch_bounds__(256) void k1_logits(const float* __restrict__ x,
                                                 const float* __restrict__ Wkf,
                                                 const float* __restrict__ bk,
                                                 float* __restrict__ L) {
    __shared__ float wlds[Kdim][Cdim + 1];      // stride 257 -> conflict-free A reads
    __shared__ float bks[Kdim];

    const int tid = threadIdx.x;
    for (int i = tid; i < Kdim * Cdim; i += 256)
        wlds[i >> 8][i & 255] = Wkf[i];
    if (tid < Kdim) bks[tid] = bk[tid];
    __syncthreads();

    const int  b      = blockIdx.y;
    const int  wave   = tid >> 5;
    const int  lane   = tid & 31;
    const int  lanelo = lane & 15;
    const int  half   = lane >> 4;
    const long n0     = (long)blockIdx.x * 512 + wave * 64;
    const float* xb   = x + (long)b * Cdim * Ndim;

    v8f aA0 = {}, aA1 = {}, aA2 = {}, aA3 = {};     // M = 0..15, N-tiles 0..3
    v8f aB0 = {}, aB1 = {}, aB2 = {}, aB3 = {};     // M = 16..31

    #pragma unroll 2
    for (int c0 = 0; c0 < Cdim; c0 += 4) {
        const int r = c0 + 2 * half;                // this lane's K pair: r, r+1
        const v4f x0 = *(const v4f*)(xb + (long)r       * Ndim + n0 + lanelo * 4);
        const v4f x1 = *(const v4f*)(xb + (long)(r + 1) * Ndim + n0 + lanelo * 4);
        v2f a0, a1;
        a0.x = wlds[lanelo][r];      a0.y = wlds[lanelo][r + 1];
        a1.x = wlds[16 + lanelo][r]; a1.y = wlds[16 + lanelo][r + 1];
        const v2f b0 = {x0.x, x1.x}, b1 = {x0.y, x1.y};
        const v2f b2 = {x0.z, x1.z}, b3 = {x0.w, x1.w};
        aA0 = WMMA_F32(a0, b0, aA0);  aA1 = WMMA_F32(a0, b1, aA1);
        aA2 = WMMA_F32(a0, b2, aA2);  aA3 = WMMA_F32(a0, b3, aA3);
        aB0 = WMMA_F32(a1, b0, aB0);  aB1 = WMMA_F32(a1, b1, aB1);
        aB2 = WMMA_F32(a1, b2, aB2);  aB3 = WMMA_F32(a1, b3, aB3);
    }

    float* Lb = L + (long)b * Kdim * Ndim;
    #pragma unroll
    for (int r = 0; r < 8; ++r) {
        const int k0 = r + 8 * half;
        const float bb0 = bks[k0], bb1 = bks[k0 + 16];
        const v4f s0 = {aA0[r] + bb0, aA1[r] + bb0, aA2[r] + bb0, aA3[r] + bb0};
        const v4f s1 = {aB0[r] + bb1, aB1[r] + bb1, aB2[r] + bb1, aB3[r] + bb1};
        *(v4f*)(Lb + (long)k0        * Ndim + n0 + lanelo * 4) = s0;
        *(v4f*)(Lb + (long)(k0 + 16) * Ndim + n0 + lanelo * 4) = s1;
    }
}

// ---------------------------------------------------------------------------
// K2: per (b,k) row (256 rows x 65536): rowmax m[] and sumexp S[].
// L (64 MB) is L2-resident (192 MB), so the second pass is cheap. float4 scans.
// ---------------------------------------------------------------------------
__global__ __launch_bounds__(256) void k2_stats(const float* __restrict__ L,
                                                float* __restrict__ m,
                                                float* __restrict__ S) {
    __shared__ float red[256];
    const int row = blockIdx.x;                 // b*32 + k
    const int tid = threadIdx.x;
    const v4f* Lr = (const v4f*)(L + (long)row * Ndim);

    float mx = -3.4e38f;
    for (int i = tid; i < Ndim / 4; i += 256) {
        const v4f v = Lr[i];
        mx = fmaxf(mx, fmaxf(fmaxf(v.x, v.y), fmaxf(v.z, v.w)));
    }
    red[tid] = mx;
    __syncthreads();
    for (int s = 128; s > 0; s >>= 1) {
        if (tid < s) red[tid] = fmaxf(red[tid], red[tid + s]);
        __syncthreads();
    }
    const float rowmax = red[0];
    __syncthreads();

    float sum = 0.f;
    for (int i = tid; i < Ndim / 4; i += 256) {
        const v4f v = Lr[i];
        sum += __expf(v.x - rowmax) + __expf(v.y - rowmax)
             + __expf(v.z - rowmax) + __expf(v.w - rowmax);
    }
    red[tid] = sum;
    __syncthreads();
    for (int s = 128; s > 0; s >>= 1) {
        if (tid < s) red[tid] += red[tid + s];
        __syncthreads();
    }
    if (tid == 0) { m[row] = rowmax; S[row] = red[0]; }
}

// ---------------------------------------------------------------------------
// K3: out[b,c,n] = sum_k Wv[c,k] * attn_norm[k,n] + x[b,c,n]
// Each wave: 64-column strip.  Build the 32x64 softmax tile in LDS (lane l
// owns columns 2l, 2l+1: b64 L-loads, exp, /S, column sums -> inv[] in LDS),
// then Wv(256x32) @ attn(32x64): 16 M-tiles x 8 K-steps x 4 N-tiles of f32
// WMMA with inv folded into the B operand; residual add fused, b128 I/O.
// ---------------------------------------------------------------------------
__global__ __launch_bounds__(256) void k3_out(const float* __restrict__ x,
                                              const float* __restrict__ L,
                                              const float* __restrict__ m,
                                              const float* __restrict__ S,
                                              const float* __restrict__ Wv,
                                              float* __restrict__ out) {
    __shared__ float wv[Cdim][Kdim + 1];        // 256x33, odd stride -> clean banks
    __shared__ float wn[8][Kdim][64];           // per-wave 32x64 softmax tile
    __shared__ float invc[8][64];               // per-wave per-column 1/(ksum+eps)

    const int tid = threadIdx.x;
    for (int i = tid; i < Cdim * Kdim; i += 256)
        wv[i >> 5][i & 31] = Wv[i];             // Wv row-major (C,K)
    __syncthreads();

    const int  b      = blockIdx.y;
    const int  wave   = tid >> 5;
    const int  lane   = tid & 31;
    const int  lanelo = lane & 15;
    const int  half   = lane >> 4;
    const long n0     = (long)blockIdx.x * 512 + wave * 64;
    const float* Lb   = L + (long)b * Kdim * Ndim;

    // ---- build normalized softmax weights; lane owns columns 2l, 2l+1 ----
    float cs0 = 0.f, cs1 = 0.f;
    #pragma unroll 4
    for (int k = 0; k < Kdim; ++k) {
        const float mk = m[b * Kdim + k];
        const float rS = 1.0f / S[b * Kdim + k];
        const v2f lv = *(const v2f*)(Lb + (long)k * Ndim + n0 + 2 * lane);
        const float e0 = __expf(lv.x - mk) * rS;
        const float e1 = __expf(lv.y - mk) * rS;
        *(v2f*)&wn[wave][k][2 * lane] = (v2f){e0, e1};
        cs0 += e0; cs1 += e1;
    }
    *(v2f*)&invc[wave][2 * lane] = (v2f){1.0f / (cs0 + 1e-9f), 1.0f / (cs1 + 1e-9f)};
    __syncthreads();                            // wn/invc visible across lanes

    const v4f inv4 = *(const v4f*)&invc[wave][lanelo * 4];   // this lane's 4 columns
    const float* xb = x   + (long)b * Cdim * Ndim;
    float*       ob = out + (long)b * Cdim * Ndim;

    for (int m0 = 0; m0 < Cdim; m0 += 16) {     // 16 output-row tiles
        v8f c0v = {}, c1v = {}, c2v = {}, c3v = {};
        #pragma unroll
        for (int k0 = 0; k0 < Kdim; k0 += 4) {  // K = 32 -> 8 WMMA steps
            const int r = k0 + 2 * half;
            v2f a;
            a.x = wv[m0 + lanelo][r];
            a.y = wv[m0 + lanelo][r + 1];
            const v4f r0 = *(const v4f*)&wn[wave][r][lanelo * 4];
            const v4f r1 = *(const v4f*)&wn[wave][r + 1][lanelo * 4];
            const v2f b0 = {r0.x * inv4.x, r1.x * inv4.x};
            const v2f b1 = {r0.y * inv4.y, r1.y * inv4.y};
            const v2f b2 = {r0.z * inv4.z, r1.z * inv4.z};
            const v2f b3 = {r0.w * inv4.w, r1.w * inv4.w};
            c0v = WMMA_F32(a, b0, c0v);  c1v = WMMA_F32(a, b1, c1v);
            c2v = WMMA_F32(a, b2, c2v);  c3v = WMMA_F32(a, b3, c3v);
        }
        #pragma unroll
        for (int r = 0; r < 8; ++r) {
            const int  c   = m0 + r + 8 * half;
            const long idx = (long)c * Ndim + n0 + lanelo * 4;
            const v4f xr = *(const v4f*)(xb + idx);
            const v4f o  = {c0v[r] + xr.x, c1v[r] + xr.y, c2v[r] + xr.z, c3v[r] + xr.w};
            *(v4f*)(ob + idx) = o;              // fused residual, 512 B/wave store
        }
    }
}

// ---------------------------------------------------------------------------
// Inputs (setup_inputs order): x, Wf, bf, Wk, Wv   (all float32)
// Workspace: L (64 MB) | Wkf (8192 f) | bk (32 f) | m (256 f) | S (256 f)
// ---------------------------------------------------------------------------
extern "C" void kernel_launch(void* const* d_in, const int* in_sizes, int n_in,
                              void* d_out, int out_size, void* d_ws, size_t ws_size,
                              hipStream_t stream) {
    const float* x  = (const float*)d_in[0];
    const float* Wf = (const float*)d_in[1];
    const float* bf = (const float*)d_in[2];
    const float* Wk = (const float*)d_in[3];
    const float* Wv = (const float*)d_in[4];
    float* out = (float*)d_out;

    float* ws   = (float*)d_ws;
    float* L    = ws;                                   // 16,777,216 floats
    float* Wkf  = ws + (size_t)Bdim * Kdim * Ndim;      // 8192
    float* bk   = Wkf + Kdim * Cdim;                    // 32
    float* mrow = bk + Kdim;                            // 256
    float* Srow = mrow + Bdim * Kdim;                   // 256

    k0_fold  <<<1, 256, 0, stream>>>(Wk, Wf, bf, Wkf, bk);
    k1_logits<<<dim3(Ndim / 512, Bdim), 256, 0, stream>>>(x, Wkf, bk, L);
    k2_stats <<<Bdim * Kdim, 256, 0, stream>>>(L, mrow, Srow);
    k3_out   <<<dim3(Ndim / 512, Bdim), 256, 0, stream>>>(x, L, mrow, Srow, Wv, out);
}